// AGBF_30545807409951
// MI455X (gfx1250) — compile-verified
//
#include <hip/hip_runtime.h>
#include <hip/hip_bf16.h>
#include <math.h>

typedef __attribute__((ext_vector_type(2))) float v2f;
typedef __attribute__((ext_vector_type(8))) float v8f;

#define HID 8
#define NP 576            // patches per batch (24*24)
#define NB 2
#define HW 384
#define PS 16
#define SCALE 0.35355339059327373f   // 8^-0.5

__device__ __forceinline__ v8f wmma_f32_4(v2f a, v2f b, v8f c) {
  // D = A(16x4,f32) * B(4x16,f32) + C(16x16,f32)  -> v_wmma_f32_16x16x4_f32
  return __builtin_amdgcn_wmma_f32_16x16x4_f32(false, a, false, b, (short)0, c,
                                               false, false);
}

__global__ void AGBF_init_kernel(unsigned int* maxbits) { *maxbits = 0u; }

// ---------------------------------------------------------------------------
// Kernel 1: patchify + QKV projection via f32 WMMA.
// grid = 72 (16-row tiles of the 1152 pat rows), block = 32 (one wave).
// Pass 1 computes [Q|K] (N columns 0-7 = Q, 8-15 = K), pass 2 computes V.
// ---------------------------------------------------------------------------
__global__ void AGBF_qkv_kernel(const float* __restrict__ x,
                                const float* __restrict__ Wq, const float* __restrict__ bq,
                                const float* __restrict__ Wk, const float* __restrict__ bk,
                                const float* __restrict__ Wv, const float* __restrict__ bv,
                                float* __restrict__ Q, float* __restrict__ K,
                                float* __restrict__ V) {
  const int tile = blockIdx.x;            // 0..71
  const int lane = threadIdx.x;           // 0..31
  const int lo   = lane & 15;
  const bool hi  = lane >= 16;
  const int ksel = hi ? 2 : 0;

  // A fragment source: pat row = tile*16 + lo
  const int rowp = tile * 16 + lo;
  const int b    = rowp / NP;
  const int n    = rowp % NP;
  const int py   = n / 24, px = n % 24;
  const float* xb = x + b * (HW * HW) + (py * PS) * HW + px * PS;

  // -------- pass 1: [Q|K] --------
  v8f acc = {0.f, 0.f, 0.f, 0.f, 0.f, 0.f, 0.f, 0.f};
  const float* WselQK = (lo < 8) ? Wq : Wk;
  const int col8 = lo & 7;
  for (int k0 = 0; k0 < 256; k0 += 4) {
    int c0 = k0 + ksel;
    v2f a;  a.x = xb[(c0 >> 4) * HW + (c0 & 15)];
            a.y = xb[((c0 + 1) >> 4) * HW + ((c0 + 1) & 15)];
    v2f bb; bb.x = WselQK[c0 * HID + col8];
            bb.y = WselQK[(c0 + 1) * HID + col8];
    acc = wmma_f32_4(a, bb, acc);
  }
#pragma unroll
  for (int r = 0; r < 8; ++r) {
    int row = tile * 16 + r + (hi ? 8 : 0);
    if (lo < 8) Q[row * HID + lo]       = acc[r] + bq[lo];
    else        K[row * HID + (lo - 8)] = acc[r] + bk[lo - 8];
  }

  // -------- pass 2: V (columns 8-15 are a duplicate, discarded) --------
  v8f accv = {0.f, 0.f, 0.f, 0.f, 0.f, 0.f, 0.f, 0.f};
  for (int k0 = 0; k0 < 256; k0 += 4) {
    int c0 = k0 + ksel;
    v2f a;  a.x = xb[(c0 >> 4) * HW + (c0 & 15)];
            a.y = xb[((c0 + 1) >> 4) * HW + ((c0 + 1) & 15)];
    v2f bb; bb.x = Wv[c0 * HID + col8];
            bb.y = Wv[(c0 + 1) * HID + col8];
    accv = wmma_f32_4(a, bb, accv);
  }
#pragma unroll
  for (int r = 0; r < 8; ++r) {
    int row = tile * 16 + r + (hi ? 8 : 0);
    if (lo < 8) V[row * HID + lo] = accv[r] + bv[lo];
  }
}

// ---------------------------------------------------------------------------
// Kernel 2: single-head attention O = softmax(Q K^T * SCALE) V, N=576, d=8.
// grid = NB*36 (batch, 16-row tile), block = 256 (8 waves).
// Scores via WMMA into LDS, softmax via 16-lane shuffle reductions + v_exp_f32
// (TRANS pipe co-executes with VALU on CDNA5), AV via VALU dot products.
// ---------------------------------------------------------------------------
__global__ void AGBF_attn_kernel(const float* __restrict__ Q,
                                 const float* __restrict__ K,
                                 const float* __restrict__ V,
                                 float* __restrict__ O) {
  __shared__ float sS[16][592];           // 16 rows x 576 scores (+pad)
  const int blk = blockIdx.x;
  const int b   = blk / 36;
  const int it  = blk % 36;
  const int tid = threadIdx.x;
  const int wv  = tid >> 5;
  const int lane = tid & 31;
  const int lo  = lane & 15;
  const bool hi = lane >= 16;
  const int ksel = hi ? 2 : 0;
  const float* Qb = Q + b * NP * HID;
  const float* Kb = K + b * NP * HID;
  const float* Vb = V + b * NP * HID;

  // phase 1: scores for this 16-row stripe (each wave covers j-tiles wv, wv+8, ...)
  const int arow = it * 16 + lo;
  v2f a0, a1;
  a0.x = Qb[arow * HID + ksel];     a0.y = Qb[arow * HID + ksel + 1];
  a1.x = Qb[arow * HID + 4 + ksel]; a1.y = Qb[arow * HID + 4 + ksel + 1];
  for (int jt = wv; jt < 36; jt += 8) {
    const int brow = jt * 16 + lo;
    v2f b0, b1;
    b0.x = Kb[brow * HID + ksel];     b0.y = Kb[brow * HID + ksel + 1];
    b1.x = Kb[brow * HID + 4 + ksel]; b1.y = Kb[brow * HID + 4 + ksel + 1];
    v8f acc = {0.f, 0.f, 0.f, 0.f, 0.f, 0.f, 0.f, 0.f};
    acc = wmma_f32_4(a0, b0, acc);
    acc = wmma_f32_4(a1, b1, acc);
#pragma unroll
    for (int r = 0; r < 8; ++r)
      sS[r + (hi ? 8 : 0)][jt * 16 + lo] = acc[r] * SCALE;
  }
  __syncthreads();

  // phase 2: row softmax (16 threads per row, shuffle reduce within half-wave)
  {
    const int r  = tid >> 4;
    const int li = tid & 15;
    float mx = -1e30f;
    for (int m = li; m < NP; m += 16) mx = fmaxf(mx, sS[r][m]);
    for (int off = 8; off > 0; off >>= 1) mx = fmaxf(mx, __shfl_xor(mx, off, 16));
    float sum = 0.f;
    for (int m = li; m < NP; m += 16) {
      float e = __expf(sS[r][m] - mx);   // v_exp_f32
      sS[r][m] = e;
      sum += e;
    }
    for (int off = 8; off > 0; off >>= 1) sum += __shfl_xor(sum, off, 16);
    const float inv = 1.0f / sum;
    for (int m = li; m < NP; m += 16) sS[r][m] *= inv;
  }
  __syncthreads();

  // phase 3: O = P @ V  (16 rows x 8 cols, K=576)
  if (tid < 128) {
    const int r = tid >> 3, j = tid & 7;
    float acc = 0.f;
    for (int m = 0; m < NP; ++m) acc += sS[r][m] * Vb[m * HID + j];
    O[(b * NP + it * 16 + r) * HID + j] = acc;
  }
}

// ---------------------------------------------------------------------------
// Kernel 3: second-stage 8x8 projections (tiny, VALU).
// ---------------------------------------------------------------------------
__global__ void AGBF_proj2_kernel(const float* __restrict__ F,
                                  const float* __restrict__ Wsq, const float* __restrict__ bsq,
                                  const float* __restrict__ Wsk, const float* __restrict__ bsk,
                                  const float* __restrict__ Wsv, const float* __restrict__ bsv,
                                  float* __restrict__ Q2, float* __restrict__ K2,
                                  float* __restrict__ V2) {
  const int t = blockIdx.x * blockDim.x + threadIdx.x;
  if (t >= NB * NP * HID) return;
  const int p = t >> 3, j = t & 7;
  const float* f = F + p * HID;
  float aq = bsq[j], ak = bsk[j], av = bsv[j];
#pragma unroll
  for (int i = 0; i < HID; ++i) {
    const float fv = f[i];
    aq += fv * Wsq[i * HID + j];
    ak += fv * Wsk[i * HID + j];
    av += fv * Wsv[i * HID + j];
  }
  Q2[t] = aq; K2[t] = ak; V2[t] = av;
}

// ---------------------------------------------------------------------------
// Kernel 4: LayerNorm + BoundedSoftplus head + global max(sx,sy) via atomic.
// Softplus kept at libm accuracy: its output feeds ceil() -> integer k.
// ---------------------------------------------------------------------------
__global__ void AGBF_head_kernel(const float* __restrict__ O,
                                 const float* __restrict__ ln_g, const float* __restrict__ ln_b,
                                 const float* __restrict__ Wp,  const float* __restrict__ bp,
                                 float* __restrict__ S, unsigned int* __restrict__ maxbits) {
  const int p = blockIdx.x * blockDim.x + threadIdx.x;
  if (p >= NB * NP) return;
  float o[HID];
  float mu = 0.f;
#pragma unroll
  for (int i = 0; i < HID; ++i) { o[i] = O[p * HID + i]; mu += o[i]; }
  mu *= 0.125f;
  float var = 0.f;
#pragma unroll
  for (int i = 0; i < HID; ++i) { float d = o[i] - mu; var += d * d; }
  var *= 0.125f;
  const float inv = rsqrtf(var + 1e-5f);
#pragma unroll
  for (int i = 0; i < HID; ++i) o[i] = (o[i] - mu) * inv * ln_g[i] + ln_b[i];
  float smax = 0.f;
#pragma unroll
  for (int j = 0; j < 3; ++j) {
    float z = bp[j];
#pragma unroll
    for (int i = 0; i < HID; ++i) z += o[i] * Wp[i * 3 + j];
    // numerically stable softplus = logaddexp(z, 0)
    const float sp = z > 0.f ? z + log1pf(expf(-z)) : log1pf(expf(z));
    const float s = fminf(sp, 6.0f) + 1e-6f;
    S[p * 3 + j] = s;
    if (j < 2) smax = fmaxf(smax, s);
  }
  atomicMax(maxbits, __float_as_uint(smax));  // all values > 0: bit order == float order
}

// ---------------------------------------------------------------------------
// Kernel 5: data-dependent kernel size computed on device (graph-capturable).
// ---------------------------------------------------------------------------
__global__ void AGBF_kparam_kernel(const unsigned int* __restrict__ maxbits,
                                   int* __restrict__ halfk) {
  const float m = __uint_as_float(*maxbits);
  int k = 2 * (int)ceilf(m + 1.0f);
  if ((k & 1) == 0) k += 1;
  *halfk = k >> 1;
}

// ---------------------------------------------------------------------------
// Kernel 6: bilateral filter with per-pixel sigmas (read at patch resolution)
// and dynamic half-width. Hot loop: one v_exp_f32 for the x-spatial term and
// one for the range term per tap (y-spatial term hoisted per row); TRANS pipe
// co-executes with the VALU accumulation. Row prefetch -> global_prefetch_b8.
// ---------------------------------------------------------------------------
__global__ void AGBF_bilateral_kernel(const float* __restrict__ x,
                                      const float* __restrict__ S,
                                      const int* __restrict__ halfk,
                                      float* __restrict__ out) {
  const int t = blockIdx.x * blockDim.x + threadIdx.x;
  if (t >= NB * HW * HW) return;
  const int b   = t / (HW * HW);
  const int rem = t - b * (HW * HW);
  const int y   = rem / HW;
  const int x0  = rem - y * HW;
  const float* xb = x + b * (HW * HW);

  const int pidx = b * NP + (y >> 4) * 24 + (x0 >> 4);
  const float sx = S[pidx * 3 + 0];
  const float sy = S[pidx * 3 + 1];
  const float sr = S[pidx * 3 + 2];
  const float isx = -0.5f / (sx * sx);
  const float isy = -0.5f / (sy * sy);
  const float isr = -0.5f / (sr * sr);
  const float c = xb[y * HW + x0];
  const int half = *halfk;

  float acc = 0.f, wsum = 0.f;
  for (int dy = -half; dy <= half; ++dy) {
    const int yy = y + dy;
    const bool yok = (yy >= 0) && (yy < HW);
    // hint the next row's stripe into cache (lowers to global_prefetch_b8)
    if (yok && dy < half && (yy + 1) < HW)
      __builtin_prefetch(xb + (yy + 1) * HW + x0, 0, 1);
    const float ey = __expf((float)(dy * dy) * isy);   // hoisted per row
    for (int dx = -half; dx <= half; ++dx) {
      const int xx = x0 + dx;
      const float v = (yok && xx >= 0 && xx < HW) ? xb[yy * HW + xx] : 0.f;
      const float d = c - v;
      const float w = ey * __expf((float)(dx * dx) * isx) * __expf(d * d * isr);
      wsum += w;
      acc  += w * v;
    }
  }
  out[t] = acc / (wsum + 1e-8f);
}

// ---------------------------------------------------------------------------
extern "C" void kernel_launch(void* const* d_in, const int* in_sizes, int n_in,
                              void* d_out, int out_size, void* d_ws, size_t ws_size,
                              hipStream_t stream) {
  const float* x    = (const float*)d_in[0];
  const float* Wq   = (const float*)d_in[1];
  const float* bq   = (const float*)d_in[2];
  const float* Wk   = (const float*)d_in[3];
  const float* bk   = (const float*)d_in[4];
  const float* Wv   = (const float*)d_in[5];
  const float* bv   = (const float*)d_in[6];
  const float* Wsq  = (const float*)d_in[7];
  const float* bsq  = (const float*)d_in[8];
  const float* Wsk  = (const float*)d_in[9];
  const float* bsk  = (const float*)d_in[10];
  const float* Wsv  = (const float*)d_in[11];
  const float* bsv  = (const float*)d_in[12];
  const float* ln_g = (const float*)d_in[13];
  const float* ln_b = (const float*)d_in[14];
  const float* Wp   = (const float*)d_in[15];
  const float* bp   = (const float*)d_in[16];

  float* ws = (float*)d_ws;
  float* Q1 = ws;
  float* K1 = ws + 9216;
  float* V1 = ws + 18432;
  float* F  = ws + 27648;
  float* Q2 = ws + 36864;
  float* K2 = ws + 46080;
  float* V2 = ws + 55296;
  float* O2 = ws + 64512;
  float* S  = ws + 73728;           // [NB*NP][3]
  unsigned int* maxbits = (unsigned int*)(ws + 77184);
  int* halfk = (int*)(ws + 77185);
  float* out = (float*)d_out;

  AGBF_init_kernel<<<1, 1, 0, stream>>>(maxbits);
  AGBF_qkv_kernel<<<72, 32, 0, stream>>>(x, Wq, bq, Wk, bk, Wv, bv, Q1, K1, V1);
  AGBF_attn_kernel<<<NB * 36, 256, 0, stream>>>(Q1, K1, V1, F);
  AGBF_proj2_kernel<<<36, 256, 0, stream>>>(F, Wsq, bsq, Wsk, bsk, Wsv, bsv, Q2, K2, V2);
  AGBF_attn_kernel<<<NB * 36, 256, 0, stream>>>(Q2, K2, V2, O2);
  AGBF_head_kernel<<<(NB * NP + 255) / 256, 256, 0, stream>>>(O2, ln_g, ln_b, Wp, bp, S, maxbits);
  AGBF_kparam_kernel<<<1, 1, 0, stream>>>(maxbits, halfk);
  AGBF_bilateral_kernel<<<(NB * HW * HW + 255) / 256, 256, 0, stream>>>(x, S, halfk, out);
}